// DecoderAttention_12893491822632
// MI455X (gfx1250) — compile-verified
//
#include <hip/hip_runtime.h>
#include <hip/hip_bf16.h>
#include <math.h>
#include <stdint.h>

// Problem constants (from reference): B=128, S=64, T=20, H=512, V=20000, L=4
#define Bc 128
#define Sc 64
#define Tc 20
#define Hc 512
#define Vc 20000
#define Lc 4
#define START_ID 1

#define WPB 8     // waves per GEMM block (256 threads)
#define KC  64    // K-chunk staged in LDS per iteration

// Async global->LDS staging via inline CDNA5 asm (tracked by ASYNCcnt).
#define USE_ASYNC_LDS 1

typedef float v2f __attribute__((ext_vector_type(2)));
typedef float v8f __attribute__((ext_vector_type(8)));

// ---------------------------------------------------------------------------
// WMMA fp32 GEMM:  C[M=128,N] = A[128,K] @ W[N,K]^T (+ bias[N])
//
// grid = (ceil((N/16/NB)/WPB), 8).  Block: 8 waves; all waves share one
// 16-row M-tile (blockIdx.y) whose A slice is staged through LDS in KC-sized
// chunks (double buffered, GLOBAL_LOAD_ASYNC_TO_LDS_B128 DMA).  Each wave
// owns NB adjacent 16x16 N-tiles; per chunk all 16 A fragments are hoisted
// from LDS into registers and the B stream is software-pipelined one k-step
// deep so V_WMMA_F32_16X16X4_F32 issue overlaps the B-fragment loads.
//
// Fragment layout (ISA 7.12.2, 32-bit):
//   A 16x4 : lanes 0-15 row M=lane {K+0,K+1}; lanes 16-31 {K+2,K+3}
//   B 4x16 : same shape with N in place of M (fed from row-major W[N,K])
//   C 16x16: VGPR i = row M = i + 8*(lane>=16), col N = lane&15
// ---------------------------------------------------------------------------
template <int NB>
__global__ __launch_bounds__(256) void wmma_gemm_atb(
        const float* __restrict__ A, const float* __restrict__ W,
        const float* __restrict__ bias, float* __restrict__ C,
        int N, int K) {
    __shared__ float aT[2][16 * KC];               // 8 KB double buffer

    const int tid  = threadIdx.x;
    const int wave = tid >> 5;
    const int lane = tid & 31;
    const int r    = lane & 15;
    const int hi   = lane >> 4;
    const int kofs = hi << 1;

    const int tm  = blockIdx.y;                    // M-tile (0..7)
    const int ntn = N >> 4;
    const int tn0 = (blockIdx.x * WPB + wave) * NB;
    const bool active = (tn0 + NB) <= ntn;         // wave-uniform

    // --- cooperative A staging: 256 threads x 16B = one 16xKC chunk ---
    const int se   = tid << 2;                     // element 0..1023
    const int srow = se / KC;
    const int scol = se % KC;
    const float* __restrict__ asrc = A + (size_t)((tm << 4) + srow) * K + scol;

    auto stage = [&](int kc, int bsel) {
#if USE_ASYNC_LDS
        const float* g = asrc + kc;
        uint32_t l = (uint32_t)(uintptr_t)(&aT[bsel][se]);
        // GLOBAL_LOAD_ASYNC_TO_LDS_B128 (GV mode): per-lane 16B memory->LDS,
        // tracked by ASYNCcnt (ISA 15.18.3 op 98).
        asm volatile("global_load_async_to_lds_b128 %0, %1, off"
                     :
                     : "v"(l), "v"(g)
                     : "memory");
#else
        float4 tmp = *(const float4*)(asrc + kc);
        *(float4*)&aT[bsel][se] = tmp;
#endif
    };

    const float* wrow[NB];
#pragma unroll
    for (int j = 0; j < NB; ++j)
        wrow[j] = active ? (W + (size_t)(((tn0 + j) << 4) + r) * K + kofs) : W;

    v8f c[NB];
#pragma unroll
    for (int j = 0; j < NB; ++j) c[j] = (v8f){0.f,0.f,0.f,0.f,0.f,0.f,0.f,0.f};

    int buf = 0;
    stage(0, buf);
    for (int kc = 0; kc < K; kc += KC) {
        const bool more = (kc + KC) < K;           // wave-uniform
        if (more) stage(kc + KC, buf ^ 1);
#if USE_ASYNC_LDS
        // current chunk's async op complete; next chunk may stay in flight
        if (more) asm volatile("s_wait_asynccnt 0x1" ::: "memory");
        else      asm volatile("s_wait_asynccnt 0x0" ::: "memory");
#endif
        __syncthreads();

        if (active) {
#pragma unroll
            for (int j = 0; j < NB; ++j)           // prefetch next W chunk
                __builtin_prefetch(wrow[j] + kc + KC, 0, 1);

            // hoist all A fragments of this chunk out of LDS
            v2f afrag[KC / 4];
#pragma unroll
            for (int kk = 0; kk < KC; kk += 4)
                afrag[kk >> 2] = *(const v2f*)&aT[buf][r * KC + kk + kofs];

            // software-pipelined B stream (1 k-step deep)
            v2f bcur[NB], bnext[NB];
#pragma unroll
            for (int j = 0; j < NB; ++j) {
                bcur[j]  = *(const v2f*)(wrow[j] + kc);
                bnext[j] = bcur[j];
            }
#pragma unroll
            for (int kk = 0; kk < KC; kk += 4) {
                if (kk + 4 < KC) {
#pragma unroll
                    for (int j = 0; j < NB; ++j)
                        bnext[j] = *(const v2f*)(wrow[j] + kc + kk + 4);
                }
#pragma unroll
                for (int j = 0; j < NB; ++j)
                    c[j] = __builtin_amdgcn_wmma_f32_16x16x4_f32(
                               false, afrag[kk >> 2], false, bcur[j],
                               (short)0, c[j], false, false);
#pragma unroll
                for (int j = 0; j < NB; ++j) bcur[j] = bnext[j];
            }
        }
        buf ^= 1;
        __syncthreads();                           // protect buffer being restaged
    }

    if (active) {
#pragma unroll
        for (int j = 0; j < NB; ++j) {
            const float bv = bias ? bias[((tn0 + j) << 4) + r] : 0.0f;
            float* __restrict__ crow =
                C + (size_t)((tm << 4) + (hi << 3)) * N + ((tn0 + j) << 4) + r;
#pragma unroll
            for (int i = 0; i < 8; ++i) crow[(size_t)i * N] = c[j][i] + bv;
        }
    }
}

// ---------------------------------------------------------------------------
// Attention: scores = q.enc, softmax, ctx = aw.enc; gathers embedding and
// assembles x0 = [emb(tok), ctx].  One 256-thread block per batch row.
// ---------------------------------------------------------------------------
__global__ void attn_ctx_kernel(const float* __restrict__ enc,   // [B,S,H]
                                const float* __restrict__ q,     // [B,H]
                                const float* __restrict__ emb,   // [V,H]
                                const int*   __restrict__ tgt,   // [B,T]
                                float* __restrict__ x0,          // [B,2H]
                                float* __restrict__ attn_out,    // [B,T,S]
                                int t) {
    __shared__ float sc[Sc];
    __shared__ float aw[Sc];
    __shared__ float red[2];

    const int b    = blockIdx.x;
    const int tid  = threadIdx.x;
    const int wave = tid >> 5;
    const int lane = tid & 31;

    const float* __restrict__ qb   = q   + (size_t)b * Hc;
    const float* __restrict__ encb = enc + (size_t)b * Sc * Hc;

    for (int s = wave; s < Sc; s += 8) {
        const float* __restrict__ es = encb + (size_t)s * Hc;
        float p = 0.f;
        for (int h = lane; h < Hc; h += 32) p += qb[h] * es[h];
        for (int off = 16; off; off >>= 1) p += __shfl_down(p, off, 32);
        if (lane == 0) sc[s] = p;
    }
    __syncthreads();

    if (tid == 0) {
        float mx = sc[0];
        for (int s = 1; s < Sc; ++s) mx = fmaxf(mx, sc[s]);
        float sum = 0.f;
        for (int s = 0; s < Sc; ++s) sum += __expf(sc[s] - mx);
        red[0] = mx;
        red[1] = 1.0f / sum;
    }
    __syncthreads();

    if (tid < Sc) {
        float a = __expf(sc[tid] - red[0]) * red[1];
        aw[tid] = a;
        attn_out[((size_t)b * Tc + t) * Sc + tid] = a;
    }
    __syncthreads();

    const int tok = (t == 0) ? START_ID : tgt[b * Tc + (t - 1)];
    const float* __restrict__ erow = emb + (size_t)tok * Hc;
    float* __restrict__ xb = x0 + (size_t)b * (2 * Hc);
    for (int h = tid; h < Hc; h += 256) {
        xb[h] = erow[h];
        float cv = 0.f;
        for (int s = 0; s < Sc; ++s) cv += aw[s] * encb[(size_t)s * Hc + h];
        xb[Hc + h] = cv;
    }
}

// ---------------------------------------------------------------------------
// GRU gate nonlinearity (torch order r,z,n):  h' = (1-z)*n + z*h
// ---------------------------------------------------------------------------
__global__ void gru_ew_kernel(const float* __restrict__ gx,
                              const float* __restrict__ gh,
                              const float* __restrict__ hprev,
                              float* __restrict__ hnew) {
    const int idx = blockIdx.x * blockDim.x + threadIdx.x;   // < B*H
    const int b = idx >> 9;
    const int h = idx & (Hc - 1);
    const float* __restrict__ gxb = gx + (size_t)b * (3 * Hc);
    const float* __restrict__ ghb = gh + (size_t)b * (3 * Hc);

    const float xr = gxb[h], xz = gxb[Hc + h], xn = gxb[2 * Hc + h];
    const float hr = ghb[h], hz = ghb[Hc + h], hn = ghb[2 * Hc + h];

    const float r = 1.f / (1.f + __expf(-(xr + hr)));
    const float z = 1.f / (1.f + __expf(-(xz + hz)));
    const float n = tanhf(xn + r * hn);
    const float hp = hprev[idx];
    hnew[idx] = (1.f - z) * n + z * hp;
}

// ---------------------------------------------------------------------------
// Row-wise log-softmax over V=20000, writing log_probs[b, t, :].
// ---------------------------------------------------------------------------
__global__ void logsoftmax_kernel(const float* __restrict__ logits, // [B,V]
                                  float* __restrict__ out,          // [B,T,V]
                                  int t) {
    __shared__ float red[256];
    const int b = blockIdx.x, tid = threadIdx.x;
    const float* __restrict__ row = logits + (size_t)b * Vc;

    float mx = -INFINITY;
    for (int v = tid; v < Vc; v += 256) mx = fmaxf(mx, row[v]);
    red[tid] = mx;
    __syncthreads();
    for (int s = 128; s; s >>= 1) {
        if (tid < s) red[tid] = fmaxf(red[tid], red[tid + s]);
        __syncthreads();
    }
    mx = red[0];
    __syncthreads();

    float sum = 0.f;
    for (int v = tid; v < Vc; v += 256) sum += __expf(row[v] - mx);
    red[tid] = sum;
    __syncthreads();
    for (int s = 128; s; s >>= 1) {
        if (tid < s) red[tid] += red[tid + s];
        __syncthreads();
    }
    const float lse = mx + __logf(red[0]);

    float* __restrict__ orow = out + ((size_t)b * Tc + t) * Vc;
    for (int v = tid; v < Vc; v += 256) orow[v] = row[v] - lse;
}

// ---------------------------------------------------------------------------
// Host-side orchestration (graph-capture safe: stream-only, async D2D copies)
// ---------------------------------------------------------------------------
extern "C" void kernel_launch(void* const* d_in, const int* in_sizes, int n_in,
                              void* d_out, int out_size, void* d_ws, size_t ws_size,
                              hipStream_t stream) {
    const float* enc  = (const float*)d_in[0];   // [B,S,H]
    const float* ehid = (const float*)d_in[1];   // [L,B,H]
    const int*   tgt  = (const int*)  d_in[2];   // [B,T]
    const float* emb  = (const float*)d_in[3];   // [V,H]
    const float* attw = (const float*)d_in[4];   // [H,H]
    const float* attb = (const float*)d_in[5];   // [H]
    const float* wih0 = (const float*)d_in[6];   // [3H,2H]
    const float* wihr = (const float*)d_in[7];   // [L-1,3H,H]
    const float* whh  = (const float*)d_in[8];   // [L,3H,H]
    const float* bih  = (const float*)d_in[9];   // [L,3H]
    const float* bhh  = (const float*)d_in[10];  // [L,3H]
    const float* outw = (const float*)d_in[11];  // [V,H]
    const float* outb = (const float*)d_in[12];  // [V]

    float* out = (float*)d_out;
    // Output layout (flat, return order): log_probs | final_h | attn
    const size_t lpOff   = 0;
    const size_t fhOff   = (size_t)Bc * Tc * Vc;            // 51,200,000
    const size_t attnOff = fhOff + (size_t)Lc * Bc * Hc;    // +262,144

    // Workspace layout (floats)  -- ~14.7 MB total
    float* ws     = (float*)d_ws;
    float* q      = ws;                                   //  65,536
    float* x0     = q      + (size_t)Bc * Hc;             // 131,072
    float* gx     = x0     + (size_t)Bc * 2 * Hc;         // 196,608
    float* gh     = gx     + (size_t)Bc * 3 * Hc;         // 196,608
    float* logits = gh     + (size_t)Bc * 3 * Hc;         // 2,560,000
    float* h_a    = logits + (size_t)Bc * Vc;             // 262,144
    float* h_b    = h_a    + (size_t)Lc * Bc * Hc;        // 262,144

    (void)hipMemcpyAsync(h_a, ehid, (size_t)Lc * Bc * Hc * sizeof(float),
                         hipMemcpyDeviceToDevice, stream);
    float* h_cur = h_a;
    float* h_nxt = h_b;

    // GEMM launcher: NB=4 when N/16 divides by 4 (512, 1536), NB=2 for 20000.
    auto gemm = [&](const float* A, const float* W, const float* bias, float* C,
                    int N, int K) {
        const int ntn = N >> 4;
        if ((ntn & 3) == 0) {
            const int groups = ntn >> 2;
            dim3 g((groups + WPB - 1) / WPB, Bc / 16);
            wmma_gemm_atb<4><<<g, 256, 0, stream>>>(A, W, bias, C, N, K);
        } else {
            const int groups = ntn >> 1;
            dim3 g((groups + WPB - 1) / WPB, Bc / 16);
            wmma_gemm_atb<2><<<g, 256, 0, stream>>>(A, W, bias, C, N, K);
        }
    };

    const size_t hStride = (size_t)Bc * Hc;
    const size_t wStride = (size_t)3 * Hc * Hc;

    for (int t = 0; t < Tc; ++t) {
        // 1) attention query: q = h_cur[L-1] @ attn_w^T + attn_b
        gemm(h_cur + (Lc - 1) * hStride, attw, attb, q, Hc, Hc);

        // 2) softmax/context + embedding gather -> x0 = [e, ctx]
        attn_ctx_kernel<<<Bc, 256, 0, stream>>>(enc, q, emb, tgt, x0,
                                                out + attnOff, t);

        // 3) GRU layer 0 (input size 2H)
        gemm(x0, wih0, bih + 0, gx, 3 * Hc, 2 * Hc);
        gemm(h_cur + 0, whh + 0, bhh + 0, gh, 3 * Hc, Hc);
        gru_ew_kernel<<<(Bc * Hc) / 256, 256, 0, stream>>>(gx, gh, h_cur, h_nxt);

        // 4) GRU layers 1..3
        for (int l = 1; l < Lc; ++l) {
            gemm(h_nxt + (size_t)(l - 1) * hStride,
                 wihr + (size_t)(l - 1) * wStride,
                 bih + (size_t)l * 3 * Hc, gx, 3 * Hc, Hc);
            gemm(h_cur + (size_t)l * hStride,
                 whh + (size_t)l * wStride,
                 bhh + (size_t)l * 3 * Hc, gh, 3 * Hc, Hc);
            gru_ew_kernel<<<(Bc * Hc) / 256, 256, 0, stream>>>(
                gx, gh, h_cur + (size_t)l * hStride, h_nxt + (size_t)l * hStride);
        }

        // 5) vocab projection + log-softmax (dominant GEMM: 8x1250 tiles)
        gemm(h_nxt + (size_t)(Lc - 1) * hStride, outw, outb, logits, Vc, Hc);
        logsoftmax_kernel<<<Bc, 256, 0, stream>>>(logits, out + lpOff, t);

        float* tmp = h_cur; h_cur = h_nxt; h_nxt = tmp;
    }

    (void)hipMemcpyAsync(out + fhOff, h_cur, (size_t)Lc * Bc * Hc * sizeof(float),
                         hipMemcpyDeviceToDevice, stream);
}